// EquivSetConv_50233937494095
// MI455X (gfx1250) — compile-verified
//
#include <hip/hip_runtime.h>

typedef __attribute__((ext_vector_type(2))) float v2f;
typedef __attribute__((ext_vector_type(8))) float v8f;

#define N_NODES_C 100000
#define N_EDGES_C 20000
#define NNZ_C     1000000
#define ALPHA_C   0.5f

// ---------------------------------------------------------------------------
// f32 WMMA wrapper: D = A(16x4) * B(4x16) + C(16x16), fp32 throughout.
// A layout: lane(0-15)=row M, vgpr0/1 = K {0,1}; lanes 16-31 = K {2,3}.
// B layout: lane(0-15)=col N, vgpr0/1 = K {0,1}; lanes 16-31 = K {2,3}.
// C/D layout: vgpr j -> row j (lanes 0-15) / row j+8 (lanes 16-31), col = lane&15.
// ---------------------------------------------------------------------------
__device__ __forceinline__ v8f wmma_f32(v2f a, v2f b, v8f c) {
  return __builtin_amdgcn_wmma_f32_16x16x4_f32(false, a, false, b, (short)0, c,
                                               false, false);
}

// ---------------------------------------------------------------------------
// Out[M x 64] = A[M x 64] @ W^T + bias,  W is [64 x 64] row-major.
// One wave computes a 16x64 output tile (4 f32 WMMA accumulators).
// ---------------------------------------------------------------------------
__global__ __launch_bounds__(256) void gemm64_bias_kernel(
    const float* __restrict__ A, const float* __restrict__ W,
    const float* __restrict__ bias, float* __restrict__ Out, int M) {
  __shared__ float Bsh[64 * 64];  // Bsh[k*64 + n] = W[n][k]
  for (int i = threadIdx.x; i < 64 * 64; i += 256) {
    const int k = i >> 6, n = i & 63;
    Bsh[i] = W[n * 64 + k];
  }
  __syncthreads();

  const int lane = threadIdx.x & 31;
  const int wave = threadIdx.x >> 5;
  const int rowBase = (blockIdx.x * 8 + wave) << 4;
  if (rowBase >= M) return;
  const int ll = lane & 15;  // row (A) / col (B,C)
  const int lh = lane >> 4;  // K half select

  const float* Arow = A + (size_t)(rowBase + ll) * 64;
  v8f acc0 = {}, acc1 = {}, acc2 = {}, acc3 = {};

#pragma unroll
  for (int k0 = 0; k0 < 64; k0 += 4) {
    const int ka = k0 + lh * 2;
    v2f a;
    a.x = Arow[ka];
    a.y = Arow[ka + 1];
    const float* B0 = &Bsh[ka * 64 + ll];
    const float* B1 = &Bsh[(ka + 1) * 64 + ll];
    v2f b;
    b.x = B0[0];  b.y = B1[0];  acc0 = wmma_f32(a, b, acc0);
    b.x = B0[16]; b.y = B1[16]; acc1 = wmma_f32(a, b, acc1);
    b.x = B0[32]; b.y = B1[32]; acc2 = wmma_f32(a, b, acc2);
    b.x = B0[48]; b.y = B1[48]; acc3 = wmma_f32(a, b, acc3);
  }

#pragma unroll
  for (int j = 0; j < 8; ++j) {
    const int r = rowBase + j + lh * 8;
    float* o = Out + (size_t)r * 64;
    o[ll]      = acc0[j] + bias[ll];
    o[16 + ll] = acc1[j] + bias[16 + ll];
    o[32 + ll] = acc2[j] + bias[32 + ll];
    o[48 + ll] = acc3[j] + bias[48 + ll];
  }
}

// ---------------------------------------------------------------------------
// Fused step 3+mix:
//   T[v] = (1-a) * ( (deg[v]*X[v] | S[v]) @ W2^T + deg[v]*b2 ) + a * X0[v]
// W2 is [64 x 128] row-major.  Virtual-concat A, K = 128.
// ---------------------------------------------------------------------------
__global__ __launch_bounds__(256) void gemm2_mix_kernel(
    const float* __restrict__ X, const float* __restrict__ X0,
    const float* __restrict__ S, const float* __restrict__ deg,
    const float* __restrict__ W2, const float* __restrict__ b2,
    float* __restrict__ T, int M) {
  __shared__ float Bsh[128 * 64];  // Bsh[k*64 + n] = W2[n][k]
  for (int i = threadIdx.x; i < 128 * 64; i += 256) {
    const int k = i >> 6, n = i & 63;
    Bsh[i] = W2[n * 128 + k];
  }
  __syncthreads();

  const int lane = threadIdx.x & 31;
  const int wave = threadIdx.x >> 5;
  const int rowBase = (blockIdx.x * 8 + wave) << 4;
  if (rowBase >= M) return;
  const int ll = lane & 15;
  const int lh = lane >> 4;

  const float dg = deg[rowBase + ll];
  const float* Xrow = X + (size_t)(rowBase + ll) * 64;
  const float* Srow = S + (size_t)(rowBase + ll) * 64;
  v8f acc0 = {}, acc1 = {}, acc2 = {}, acc3 = {};

#pragma unroll
  for (int k0 = 0; k0 < 64; k0 += 4) {  // first half of K: deg*X
    const int ka = k0 + lh * 2;
    v2f a;
    a.x = dg * Xrow[ka];
    a.y = dg * Xrow[ka + 1];
    const float* B0 = &Bsh[ka * 64 + ll];
    const float* B1 = &Bsh[(ka + 1) * 64 + ll];
    v2f b;
    b.x = B0[0];  b.y = B1[0];  acc0 = wmma_f32(a, b, acc0);
    b.x = B0[16]; b.y = B1[16]; acc1 = wmma_f32(a, b, acc1);
    b.x = B0[32]; b.y = B1[32]; acc2 = wmma_f32(a, b, acc2);
    b.x = B0[48]; b.y = B1[48]; acc3 = wmma_f32(a, b, acc3);
  }
#pragma unroll
  for (int k0 = 64; k0 < 128; k0 += 4) {  // second half of K: S
    const int ka = k0 + lh * 2;
    v2f a;
    a.x = Srow[ka - 64];
    a.y = Srow[ka - 63];
    const float* B0 = &Bsh[ka * 64 + ll];
    const float* B1 = &Bsh[(ka + 1) * 64 + ll];
    v2f b;
    b.x = B0[0];  b.y = B1[0];  acc0 = wmma_f32(a, b, acc0);
    b.x = B0[16]; b.y = B1[16]; acc1 = wmma_f32(a, b, acc1);
    b.x = B0[32]; b.y = B1[32]; acc2 = wmma_f32(a, b, acc2);
    b.x = B0[48]; b.y = B1[48]; acc3 = wmma_f32(a, b, acc3);
  }

  const float oma = 1.0f - ALPHA_C;
#pragma unroll
  for (int j = 0; j < 8; ++j) {
    const int r = rowBase + j + lh * 8;
    const float dr = deg[r];
    const float* x0r = X0 + (size_t)r * 64;
    float* o = T + (size_t)r * 64;
    o[ll]      = oma * (acc0[j] + dr * b2[ll])      + ALPHA_C * x0r[ll];
    o[16 + ll] = oma * (acc1[j] + dr * b2[16 + ll]) + ALPHA_C * x0r[16 + ll];
    o[32 + ll] = oma * (acc2[j] + dr * b2[32 + ll]) + ALPHA_C * x0r[32 + ll];
    o[48 + ll] = oma * (acc3[j] + dr * b2[48 + ll]) + ALPHA_C * x0r[48 + ll];
  }
}

// ---------------------------------------------------------------------------
// Xe[edges[e]] += Y1[vertex[e]] ; deg[vertex[e]] += 1.  16 threads / incidence.
// ---------------------------------------------------------------------------
__global__ __launch_bounds__(256) void scatter_edges_kernel(
    const float* __restrict__ Y1, const int* __restrict__ vertex,
    const int* __restrict__ edges, float* __restrict__ Xe,
    float* __restrict__ deg) {
  const int tid = blockIdx.x * 256 + threadIdx.x;
  const int e = tid >> 4;
  if (e >= NNZ_C) return;
  const int c = (tid & 15) << 2;
  const int v = vertex[e];
  const int h = edges[e];
  const float4 y = *(const float4*)(Y1 + (size_t)v * 64 + c);
  float* dst = Xe + (size_t)h * 64 + c;
  unsafeAtomicAdd(dst + 0, y.x);
  unsafeAtomicAdd(dst + 1, y.y);
  unsafeAtomicAdd(dst + 2, y.z);
  unsafeAtomicAdd(dst + 3, y.w);
  if ((tid & 15) == 0) unsafeAtomicAdd(deg + v, 1.0f);
}

// ---------------------------------------------------------------------------
// S[vertex[e]] += Xe[edges[e]].
// ---------------------------------------------------------------------------
__global__ __launch_bounds__(256) void scatter_nodes_kernel(
    const float* __restrict__ Xe, const int* __restrict__ vertex,
    const int* __restrict__ edges, float* __restrict__ S) {
  const int tid = blockIdx.x * 256 + threadIdx.x;
  const int e = tid >> 4;
  if (e >= NNZ_C) return;
  const int c = (tid & 15) << 2;
  const int v = vertex[e];
  const int h = edges[e];
  const float4 y = *(const float4*)(Xe + (size_t)h * 64 + c);
  float* dst = S + (size_t)v * 64 + c;
  unsafeAtomicAdd(dst + 0, y.x);
  unsafeAtomicAdd(dst + 1, y.y);
  unsafeAtomicAdd(dst + 2, y.z);
  unsafeAtomicAdd(dst + 3, y.w);
}

extern "C" void kernel_launch(void* const* d_in, const int* in_sizes, int n_in,
                              void* d_out, int out_size, void* d_ws,
                              size_t ws_size, hipStream_t stream) {
  (void)in_sizes; (void)n_in; (void)out_size; (void)ws_size;
  const float* X    = (const float*)d_in[0];
  const float* X0   = (const float*)d_in[1];
  const int*   vertex = (const int*)d_in[2];
  const int*   edges  = (const int*)d_in[3];
  const float* W1w  = (const float*)d_in[4];
  const float* W1b  = (const float*)d_in[5];
  const float* W2w  = (const float*)d_in[6];
  const float* W2b  = (const float*)d_in[7];
  const float* Ww   = (const float*)d_in[8];
  const float* Wb   = (const float*)d_in[9];

  float* out = (float*)d_out;                       // [100000 x 64]
  float* Xe  = out + (size_t)N_NODES_C * 64;        // [20000 x 64] (output #2)

  float* ws  = (float*)d_ws;
  float* Y1  = ws;                                  // [100000 x 64], reused as T
  float* S   = Y1 + (size_t)N_NODES_C * 64;         // [100000 x 64]
  float* deg = S + (size_t)N_NODES_C * 64;          // [100000]

  // Zero accumulation targets (poisoned 0xAA and not re-zeroed between replays).
  hipMemsetAsync(Xe, 0, (size_t)N_EDGES_C * 64 * sizeof(float), stream);
  hipMemsetAsync(S, 0, ((size_t)N_NODES_C * 64 + N_NODES_C) * sizeof(float),
                 stream);  // S followed contiguously by deg

  const int tiles = N_NODES_C / 16;          // 6250 row tiles
  const int gemmBlocks = (tiles + 7) / 8;    // 8 waves (tiles) per block
  const int scatterBlocks = (NNZ_C * 16) / 256;

  // 1) Y1 = X @ W1^T + b1
  gemm64_bias_kernel<<<gemmBlocks, 256, 0, stream>>>(X, W1w, W1b, Y1, N_NODES_C);
  // 2) Xe = segment_sum(Y1[vertex], edges) ; deg = histogram(vertex)
  scatter_edges_kernel<<<scatterBlocks, 256, 0, stream>>>(Y1, vertex, edges, Xe, deg);
  // 3) S = segment_sum(Xe[edges], vertex)
  scatter_nodes_kernel<<<scatterBlocks, 256, 0, stream>>>(Xe, vertex, edges, S);
  // 4) T = (1-a)*((deg*X | S) @ W2^T + deg*b2) + a*X0   (T reuses Y1 storage)
  gemm2_mix_kernel<<<gemmBlocks, 256, 0, stream>>>(X, X0, S, deg, W2w, W2b, Y1,
                                                   N_NODES_C);
  // 5) out = T @ W^T + b
  gemm64_bias_kernel<<<gemmBlocks, 256, 0, stream>>>(Y1, Ww, Wb, out, N_NODES_C);
}